// A3TGCNCat1_87986700026277
// MI455X (gfx1250) — compile-verified
//
#include <hip/hip_runtime.h>
#include <hip/hip_bf16.h>

typedef __attribute__((ext_vector_type(2))) float v2f;
typedef __attribute__((ext_vector_type(8))) float v8f;

#define NB      32      // batch
#define NROWS   1024    // 2*512 nodes (ad rows 0..511, dis rows 512..1023)
#define NNODES  512
#define NCOLS   8
#define FDIM    128     // NCOLS*16
#define HID     64
#define NEDGES  8192
#define NPER    37

__device__ __forceinline__ void fma4(float4& a, float w, const float4 b) {
    a.x = fmaf(w, b.x, a.x); a.y = fmaf(w, b.y, a.y);
    a.z = fmaf(w, b.z, a.z); a.w = fmaf(w, b.w, a.w);
}
__device__ __forceinline__ float sigm(float x) { return 1.0f / (1.0f + __expf(-x)); }

// ---------------------------------------------------------------------------
// K1: embedding gather + conv GEMM for gates z and h (r gate is dead code).
// Hg[b, row, c] = x_emb[b,row,:] @ Wg   (row in [0,1024): ad rows then dis rows)
// Block: 128 thr (4 waves). Grid: (64 row-tiles, 32 batches).
// ---------------------------------------------------------------------------
__global__ void k_embed_conv(const int* __restrict__ xb,      // (B,1024,8)
                             const float* __restrict__ emb,   // (8,31,16)
                             const float* __restrict__ Wz,    // (128,64)
                             const float* __restrict__ Wh,    // (128,64)
                             float* __restrict__ Hz,          // (B,1024,64)
                             float* __restrict__ Hh)
{
    __shared__ float As[16 * 132];   // 16 rows x 128, stride 132 (bank-safe)
    const int b    = blockIdx.y;
    const int tile = blockIdx.x;     // 0..63
    const int t    = threadIdx.x;    // 0..127

    {   // cooperative gather: thread -> (row, embedding column)
        const int r = t >> 3, col = t & 7;
        const int row_g = tile * 16 + r;
        const int idx   = xb[(b * NROWS + row_g) * NCOLS + col];
        const float4* s = (const float4*)(emb + ((size_t)(col * 31 + idx) << 4));
        float4* d = (float4*)(&As[r * 132 + col * 16]);
        d[0] = s[0]; d[1] = s[1]; d[2] = s[2]; d[3] = s[3];
    }
    __syncthreads();

    const int lane = t & 31, wid = t >> 5;
    const int lm = lane & 15, lh = lane >> 4;
    const int ccol = wid * 16 + lm;          // output channel 0..63

    v8f accZ = {}; v8f accH = {};
    #pragma unroll 4
    for (int kk = 0; kk < 32; ++kk) {
        const int k0 = kk * 4 + 2 * lh;      // K index for this lane half
        v2f a;  a.x  = As[lm * 132 + k0];        a.y  = As[lm * 132 + k0 + 1];
        v2f bz; bz.x = Wz[k0 * HID + ccol];      bz.y = Wz[(k0 + 1) * HID + ccol];
        v2f bh; bh.x = Wh[k0 * HID + ccol];      bh.y = Wh[(k0 + 1) * HID + ccol];
        accZ = __builtin_amdgcn_wmma_f32_16x16x4_f32(false, a, false, bz, (short)0, accZ, false, false);
        accH = __builtin_amdgcn_wmma_f32_16x16x4_f32(false, a, false, bh, (short)0, accH, false, false);
    }

    const size_t rowbase = (size_t)b * NROWS + tile * 16;
    #pragma unroll
    for (int v = 0; v < 8; ++v) {
        const int M = v + 8 * lh;            // D layout: VGPR v -> rows v / v+8
        Hz[(rowbase + M) * HID + ccol] = accZ[v];
        Hh[(rowbase + M) * HID + ccol] = accH[v];
    }
}

// ---------------------------------------------------------------------------
// K2a: degree + dinv + CSR row starts (1 block, 512 threads)
// ---------------------------------------------------------------------------
__global__ void k_degree(const int* __restrict__ edges,   // src[0..8192), dst[8192..)
                         float* __restrict__ dinv,
                         int* __restrict__ rowstart,      // [513]
                         int* __restrict__ cursor)        // [512]
{
    __shared__ int cnt[512];
    __shared__ int scan[512];
    const int t = threadIdx.x;
    cnt[t] = 0;
    __syncthreads();
    for (int e = t; e < NEDGES; e += 512) atomicAdd(&cnt[edges[NEDGES + e]], 1);
    __syncthreads();
    const int c = cnt[t];
    dinv[t] = rsqrtf((float)(c + 1));
    scan[t] = c;
    __syncthreads();
    for (int off = 1; off < 512; off <<= 1) {
        int vprev = (t >= off) ? scan[t - off] : 0;
        __syncthreads();
        scan[t] += vprev;
        __syncthreads();
    }
    const int excl = scan[t] - c;
    rowstart[t] = excl;
    cursor[t]   = excl;
    if (t == 511) rowstart[512] = scan[511];
}

// K2b: scatter edges into CSR order
__global__ void k_fill(const int* __restrict__ edges, int* __restrict__ cursor,
                       int* __restrict__ srcs)
{
    const int e = blockIdx.x * 256 + threadIdx.x;
    if (e < NEDGES) {
        const int d = edges[NEDGES + e];
        const int pos = atomicAdd(&cursor[d], 1);
        srcs[pos] = edges[e];
    }
}

// ---------------------------------------------------------------------------
// K3: softmax(attention) -> pa/pd/pc per batch; bias-only constant term; zero
// the pooled accumulator. 1 block, 64 threads.
// ---------------------------------------------------------------------------
__global__ void k_prep(const float* __restrict__ attention, const int* __restrict__ LOS,
                       const float* __restrict__ cbz, const float* __restrict__ cbh,
                       const float* __restrict__ Wlz, const float* __restrict__ Wlh,
                       const float* __restrict__ lbz, const float* __restrict__ lbh,
                       float* __restrict__ pa, float* __restrict__ pd, float* __restrict__ pc,
                       float* __restrict__ hconst, float* __restrict__ pooled)
{
    __shared__ float p[NPER];
    const int t = threadIdx.x;                 // 64 threads
    float m = -1e30f;
    for (int i = 0; i < NPER; ++i) m = fmaxf(m, attention[i]);
    if (t < NPER) p[t] = __expf(attention[t] - m);
    __syncthreads();
    float s = 0.f;
    for (int i = 0; i < NPER; ++i) s += p[i];
    __syncthreads();
    if (t < NPER) p[t] = p[t] / s;
    __syncthreads();

    if (t < NB) {
        const int los = LOS[t];                // in [1,37]
        float a = 0.f;
        for (int i = 0; i + 1 < los; ++i) a += p[i];
        float cc = 0.f;
        for (int i = los; i < NPER; ++i) cc += p[i];
        pa[t] = a * (1.0f / 512.0f);           // node-mean folded into row weight
        pd[t] = p[los - 1] * (1.0f / 512.0f);
        pc[t] = cc;                            // constant term: not node-averaged
    }
    {   // H_per for t >= los: conv output is just conv bias (x == 0)
        float u = lbz[t], vv = lbh[t];
        for (int c2 = 0; c2 < HID; ++c2) {
            u  += cbz[c2] * Wlz[c2 * HID + t];
            vv += cbh[c2] * Wlh[c2 * HID + t];
        }
        hconst[t] = (1.f - sigm(u)) * tanhf(vv);
    }
    for (int i = t; i < NB * HID; i += 64) pooled[i] = 0.f;
}

// ---------------------------------------------------------------------------
// K4: CSR graph aggregation (both gates) -> LDS, WMMA vs lin_w[:64], fused
// sigmoid/tanh gate, attention-weighted node-mean reduction into pooled.
// Block: 128 thr. Grid: (64 row-tiles, 32 batches).
// ---------------------------------------------------------------------------
__global__ void k_gcn_lin(const float* __restrict__ Hz, const float* __restrict__ Hh,
                          const float* __restrict__ dinv,
                          const int* __restrict__ rowstart, const int* __restrict__ srcs,
                          const float* __restrict__ cbz, const float* __restrict__ cbh,
                          const float* __restrict__ Wlz, const float* __restrict__ Wlh,
                          const float* __restrict__ lbz, const float* __restrict__ lbh,
                          const float* __restrict__ pa, const float* __restrict__ pd,
                          float* __restrict__ pooled)
{
    __shared__ float Az[16 * 68];
    __shared__ float Ah[16 * 68];
    const int b = blockIdx.y, tile = blockIdx.x;
    const int t = threadIdx.x;

    {   // aggregation: thread -> (row r, channel octet q)
        const int r = t >> 3, q = t & 7;
        const int row_g = tile * 16 + r;
        const int part  = row_g >> 9;          // 0 = ad, 1 = dis
        const int n     = row_g & 511;
        const float dn  = dinv[n];
        const size_t selfoff = ((size_t)(b * NROWS + row_g)) * HID + q * 8;
        float4 z0 = ((const float4*)(Hz + selfoff))[0];
        float4 z1 = ((const float4*)(Hz + selfoff))[1];
        float4 h0 = ((const float4*)(Hh + selfoff))[0];
        float4 h1 = ((const float4*)(Hh + selfoff))[1];
        const float w2 = dn * dn;
        float4 az0 = {w2*z0.x, w2*z0.y, w2*z0.z, w2*z0.w};
        float4 az1 = {w2*z1.x, w2*z1.y, w2*z1.z, w2*z1.w};
        float4 ah0 = {w2*h0.x, w2*h0.y, w2*h0.z, w2*h0.w};
        float4 ah1 = {w2*h1.x, w2*h1.y, w2*h1.z, w2*h1.w};
        const int j0 = rowstart[n], j1 = rowstart[n + 1];
        const size_t pbase = (size_t)(b * NROWS + part * NNODES) * HID;
        for (int j = j0; j < j1; ++j) {
            const int s  = srcs[j];
            const float w = dn * dinv[s];
            const float4* zr = (const float4*)(Hz + pbase + (size_t)s * HID + q * 8);
            const float4* hr = (const float4*)(Hh + pbase + (size_t)s * HID + q * 8);
            fma4(az0, w, zr[0]); fma4(az1, w, zr[1]);
            fma4(ah0, w, hr[0]); fma4(ah1, w, hr[1]);
        }
        const float4 bz0 = ((const float4*)cbz)[q * 2], bz1 = ((const float4*)cbz)[q * 2 + 1];
        const float4 bh0 = ((const float4*)cbh)[q * 2], bh1 = ((const float4*)cbh)[q * 2 + 1];
        float4* dz = (float4*)(&Az[r * 68 + q * 8]);
        float4* dh = (float4*)(&Ah[r * 68 + q * 8]);
        dz[0] = {az0.x + bz0.x, az0.y + bz0.y, az0.z + bz0.z, az0.w + bz0.w};
        dz[1] = {az1.x + bz1.x, az1.y + bz1.y, az1.z + bz1.z, az1.w + bz1.w};
        dh[0] = {ah0.x + bh0.x, ah0.y + bh0.y, ah0.z + bh0.z, ah0.w + bh0.w};
        dh[1] = {ah1.x + bh1.x, ah1.y + bh1.y, ah1.z + bh1.z, ah1.w + bh1.w};
    }
    __syncthreads();

    const int lane = t & 31, wid = t >> 5;
    const int lm = lane & 15, lh = lane >> 4;
    const int ccol = wid * 16 + lm;

    v8f accU = {}; v8f accV = {};
    #pragma unroll 4
    for (int kk = 0; kk < 16; ++kk) {
        const int k0 = kk * 4 + 2 * lh;
        v2f aZ; aZ.x = Az[lm * 68 + k0];  aZ.y = Az[lm * 68 + k0 + 1];
        v2f aH; aH.x = Ah[lm * 68 + k0];  aH.y = Ah[lm * 68 + k0 + 1];
        v2f bU; bU.x = Wlz[k0 * HID + ccol]; bU.y = Wlz[(k0 + 1) * HID + ccol];
        v2f bV; bV.x = Wlh[k0 * HID + ccol]; bV.y = Wlh[(k0 + 1) * HID + ccol];
        accU = __builtin_amdgcn_wmma_f32_16x16x4_f32(false, aZ, false, bU, (short)0, accU, false, false);
        accV = __builtin_amdgcn_wmma_f32_16x16x4_f32(false, aH, false, bV, (short)0, accV, false, false);
    }

    const float bzk = lbz[ccol], bhk = lbh[ccol];
    const float w = (tile >= 32) ? pd[b] : pa[b];   // whole tile is same part
    float partial = 0.f;
    #pragma unroll
    for (int v = 0; v < 8; ++v) {
        const float u  = accU[v] + bzk;
        const float vv = accV[v] + bhk;
        partial += (1.f - sigm(u)) * tanhf(vv);
    }
    partial *= w;
    partial += __shfl_xor(partial, 16, 32);         // combine the two row halves
    if (lane < 16) atomicAdd(&pooled[b * HID + ccol], partial);
}

// ---------------------------------------------------------------------------
// K5: pooled (+ constant tail term) -> relu MLP -> (32,2) output. 1 block.
// ---------------------------------------------------------------------------
__global__ void k_cls(const float* __restrict__ pooled, const float* __restrict__ pc,
                      const float* __restrict__ hconst,
                      const float* __restrict__ W1, const float* __restrict__ b1, // (64,128),(128)
                      const float* __restrict__ W2, const float* __restrict__ b2, // (128,2),(2)
                      float* __restrict__ out)
{
    __shared__ float pl[NB * HID];
    __shared__ float h1[NB * 128];
    const int t = threadIdx.x;  // 256
    for (int i = t; i < NB * HID; i += 256) {
        const int bb = i >> 6, k = i & 63;
        pl[i] = pooled[i] + pc[bb] * hconst[k];
    }
    __syncthreads();
    for (int i = t; i < NB * 128; i += 256) {
        const int bb = i >> 7, j = i & 127;
        float acc = b1[j];
        const float* pr = &pl[bb * HID];
        for (int k = 0; k < HID; ++k) acc = fmaf(pr[k], W1[k * 128 + j], acc);
        h1[i] = fmaxf(acc, 0.f);
    }
    __syncthreads();
    if (t < NB * 2) {
        const int bb = t >> 1, j = t & 1;
        float acc = b2[j];
        const float* hr = &h1[bb * 128];
        for (int k = 0; k < 128; ++k) acc = fmaf(hr[k], W2[k * 2 + j], acc);
        out[bb * 2 + j] = acc;
    }
}

// ---------------------------------------------------------------------------
extern "C" void kernel_launch(void* const* d_in, const int* in_sizes, int n_in,
                              void* d_out, int out_size, void* d_ws, size_t ws_size,
                              hipStream_t stream)
{
    // setup_inputs() dict order:
    const int*   xb    = (const int*)  d_in[0];   // x_batch (32,1024,8)
    const int*   LOS   = (const int*)  d_in[1];   // LOS_batch (32)
    // d_in[2]=ad_col_index (arange), d_in[3]=dis_col_index (512+arange): identity, unused
    const int*   edges = (const int*)  d_in[4];   // (2,8192)
    const float* emb   = (const float*)d_in[5];   // (8,31,16)
    const float* cwz   = (const float*)d_in[6];   // conv_w_z (128,64)
    const float* cbz   = (const float*)d_in[7];
    const float* lwz   = (const float*)d_in[8];   // lin_w_z (128,64) -> rows [:64] used
    const float* lbz   = (const float*)d_in[9];
    // d_in[10..13] = r gate: dead code in reference
    const float* cwh   = (const float*)d_in[14];
    const float* cbh   = (const float*)d_in[15];
    const float* lwh   = (const float*)d_in[16];
    const float* lbh   = (const float*)d_in[17];
    const float* att   = (const float*)d_in[18];  // (37)
    const float* W1    = (const float*)d_in[19];
    const float* b1    = (const float*)d_in[20];
    const float* W2    = (const float*)d_in[21];
    const float* b2    = (const float*)d_in[22];
    float* out = (float*)d_out;

    char* ws = (char*)d_ws;
    const size_t HSZ = (size_t)NB * NROWS * HID * sizeof(float);   // 8 MB each
    float* Hz       = (float*)(ws);
    float* Hh       = (float*)(ws + HSZ);
    float* dinv     = (float*)(ws + 2 * HSZ);
    int*   rowstart = (int*)  (ws + 2 * HSZ + 4096);
    int*   cursor   = (int*)  (ws + 2 * HSZ + 8192);
    int*   srcs     = (int*)  (ws + 2 * HSZ + 12288);
    float* pa       = (float*)(ws + 2 * HSZ + 49152);
    float* pd       = (float*)(ws + 2 * HSZ + 49152 + 256);
    float* pc       = (float*)(ws + 2 * HSZ + 49152 + 512);
    float* hconst   = (float*)(ws + 2 * HSZ + 49152 + 768);
    float* pooled   = (float*)(ws + 2 * HSZ + 49152 + 1280);

    dim3 gridT(64, NB);
    k_embed_conv<<<gridT, 128, 0, stream>>>(xb, emb, cwz, cwh, Hz, Hh);
    k_degree    <<<1, 512, 0, stream>>>(edges, dinv, rowstart, cursor);
    k_fill      <<<NEDGES / 256, 256, 0, stream>>>(edges, cursor, srcs);
    k_prep      <<<1, 64, 0, stream>>>(att, LOS, cbz, cbh, lwz, lwh, lbz, lbh,
                                       pa, pd, pc, hconst, pooled);
    k_gcn_lin   <<<gridT, 128, 0, stream>>>(Hz, Hh, dinv, rowstart, srcs,
                                            cbz, cbh, lwz, lwh, lbz, lbh,
                                            pa, pd, pooled);
    k_cls       <<<1, 256, 0, stream>>>(pooled, pc, hconst, W1, b1, W2, b2, out);
    (void)in_sizes; (void)n_in; (void)out_size; (void)ws_size;
}